// Block_22686017258154
// MI455X (gfx1250) — compile-verified
//
#include <hip/hip_runtime.h>
#include <math.h>

typedef float v2f __attribute__((ext_vector_type(2)));
typedef float v8f __attribute__((ext_vector_type(8)));

#define DIM   768
#define NTOK  4096      // B*N = 4*1024
#define NHEAD 12
#define HD    64
#define HID   3072
#define QKVN  2304
#define SEQ   1024
#define QK_SCALE 0.125f // 64^-0.5

// ---------------------------------------------------------------------------
// f32 WMMA 16x16x4 wrapper (D = A(16x4) * B(4x16) + C)
// A frag: lane L holds A[L&15][2*(L>>4)+v] in v2f  (2 contiguous K elements)
// B frag: lane L holds B[2*(L>>4)+v][L&15]
// C/D:    lane L, vgpr v -> C[v + 8*(L>>4)][L&15]
// ---------------------------------------------------------------------------
__device__ __forceinline__ v8f wmma4(v2f a, v2f b, v8f c) {
  return __builtin_amdgcn_wmma_f32_16x16x4_f32(false, a, false, b, (short)0, c,
                                               false, false);
}

__device__ __forceinline__ float gelu_exact(float x) {
  return 0.5f * x * (1.0f + erff(x * 0.70710678f));
}

// ---------------------------------------------------------------------------
// Pair transform for doubled-real complex contraction:
//   out1[j] = (re, -im)   (contracts against (ar, ai) -> real part)
//   out2[j] = (im,  re)   (contracts against (ar, ai) -> imag part)
// ---------------------------------------------------------------------------
__global__ void make_w12(const float* __restrict__ w, float* __restrict__ w1,
                         float* __restrict__ w2, int n /* complex elems */) {
  const int i = blockIdx.x * 256 + threadIdx.x;
  if (i < n) {
    const float2 c = ((const float2*)w)[i];
    float2 a; a.x = c.x; a.y = -c.y;
    float2 b; b.x = c.y; b.y = c.x;
    ((float2*)w1)[i] = a;
    ((float2*)w2)[i] = b;
  }
}

// Same transform applied to the K section of the qkv buffer (strided rows).
// k1/k2: [4096, 768] complex (head-major cols, same as qkv K-section).
__global__ void make_k12(const float* __restrict__ qkv, float* __restrict__ k1,
                         float* __restrict__ k2) {
  const int i = blockIdx.x * 256 + threadIdx.x;   // over 4096*768 complex
  const int t = i / DIM, c = i % DIM;
  const float2 v = ((const float2*)qkv)[(size_t)t * QKVN + DIM + c];
  float2 a; a.x = v.x; a.y = -v.y;
  float2 b; b.x = v.y; b.y = v.x;
  ((float2*)k1)[i] = a;
  ((float2*)k2)[i] = b;
}

// ---------------------------------------------------------------------------
// Complex LayerNorm: one wave32 per token row of 768 complex values.
// ---------------------------------------------------------------------------
__global__ void ln_cplx(const float* __restrict__ x, const float* __restrict__ w,
                        const float* __restrict__ b, float* __restrict__ out) {
  const int row = blockIdx.x;
  const int lane = threadIdx.x;
  const float2* xr = (const float2*)x + (size_t)row * DIM;
  const float2* w2 = (const float2*)w;
  const float2* b2 = (const float2*)b;
  float2* o2 = (float2*)out + (size_t)row * DIM;

  float2 v[24];
  float sr = 0.f, si = 0.f;
#pragma unroll
  for (int i = 0; i < 24; ++i) {
    v[i] = xr[lane + i * 32];
    sr += v[i].x; si += v[i].y;
  }
#pragma unroll
  for (int off = 16; off >= 1; off >>= 1) {
    sr += __shfl_xor(sr, off);
    si += __shfl_xor(si, off);
  }
  const float mur = sr * (1.0f / DIM), mui = si * (1.0f / DIM);

  float var = 0.f;
#pragma unroll
  for (int i = 0; i < 24; ++i) {
    float dr = v[i].x - mur, di = v[i].y - mui;
    var += dr * dr + di * di;
  }
#pragma unroll
  for (int off = 16; off >= 1; off >>= 1) var += __shfl_xor(var, off);
  const float inv = rsqrtf(var * (1.0f / DIM) + 1e-5f);

#pragma unroll
  for (int i = 0; i < 24; ++i) {
    const int c = lane + i * 32;
    const float dr = (v[i].x - mur) * inv, di = (v[i].y - mui) * inv;
    const float2 wc = w2[c], bc = b2[c];
    float2 y;
    y.x = dr * wc.x - di * wc.y + bc.x;
    y.y = dr * wc.y + di * wc.x + bc.y;
    o2[c] = y;
  }
}

// ---------------------------------------------------------------------------
// Doubled-real complex GEMM: accr = A.W1^T, acci = A.W2^T over K2 = 2K reals.
// Every fragment is a direct aligned b64 load: 3 loads + 2 WMMAs per step.
// 256 threads = 8 waves (2x4) -> block tile 32(M) x 64(N).
// ---------------------------------------------------------------------------
template <bool HAS_BIAS, bool DO_GELU, bool HAS_RES>
__global__ void gemm_cplx2(const float* __restrict__ A,
                           const float* __restrict__ W1,
                           const float* __restrict__ W2,
                           const float* __restrict__ bias,
                           const float* __restrict__ res,
                           float* __restrict__ out, int M, int N, int K2) {
  const int lane = threadIdx.x & 31;
  const int wave = threadIdx.x >> 5;
  const int lo = lane & 15, hi = lane >> 4;
  const int koff = hi * 2;
  const int tileM = blockIdx.y * 32 + (wave >> 2) * 16;
  const int tileN = blockIdx.x * 64 + (wave & 3) * 16;

  const float* Ap  = A  + (size_t)(tileM + lo) * K2 + koff;
  const float* W1p = W1 + (size_t)(tileN + lo) * K2 + koff;
  const float* W2p = W2 + (size_t)(tileN + lo) * K2 + koff;

  const v8f z = {0, 0, 0, 0, 0, 0, 0, 0};
  v8f accr = z, acci = z;

  for (int k0 = 0; k0 < K2; k0 += 4) {
    const v2f a  = *(const v2f*)(Ap + k0);
    const v2f b1 = *(const v2f*)(W1p + k0);
    const v2f b2 = *(const v2f*)(W2p + k0);
    accr = wmma4(a, b1, accr);
    acci = wmma4(a, b2, acci);
  }

  const float2* bias2 = (const float2*)bias;
  const float2* res2 = (const float2*)res;
  float2* out2 = (float2*)out;
#pragma unroll
  for (int v = 0; v < 8; ++v) {
    const int m = tileM + v + 8 * hi;
    const int n = tileN + lo;
    float re = accr[v], im = acci[v];
    if (HAS_BIAS) { const float2 bb = bias2[n]; re += bb.x; im += bb.y; }
    if (DO_GELU)  { re = gelu_exact(re); im = gelu_exact(im); }
    if (HAS_RES)  { const float2 rr = res2[(size_t)m * N + n]; re += rr.x; im += rr.y; }
    float2 o; o.x = re; o.y = im;
    out2[(size_t)m * N + n] = o;
  }
}

// ---------------------------------------------------------------------------
// Flash attention: one wave per (b, head, 16-query tile).
// qkv: [4096, 2304] complex rows (Q/K/V sections); scores use the doubled-real
// trick with pre-transformed K1=(kr,-ki), K2=(ki,kr): interleaved Q in LDS is
// already the doubled-real row, so the scores loop is pure loads + WMMAs.
// ctx: [4096, 768] complex.
// ---------------------------------------------------------------------------
__global__ void attn_flash(const float* __restrict__ qkv,
                           const float* __restrict__ k1,
                           const float* __restrict__ k2,
                           float* __restrict__ ctx) {
  const int lane = threadIdx.x;
  const int lo = lane & 15, hi = lane >> 4;
  const int koff = hi * 2;
  const int bh = blockIdx.x;
  const int b = bh / NHEAD, h = bh % NHEAD;
  const int qt = blockIdx.y;

  __shared__ float qs[16 * 2 * HD]; // Q tile (pre-scaled), [16][128] doubled-real
  __shared__ float ps[16 * 16];     // softmax probs tile

  const float2* qkv2 = (const float2*)qkv;
  const size_t base = (size_t)b * SEQ * QKVN;
  const size_t btok = (size_t)b * SEQ;

  // stage Q tile into LDS, pre-scaled by 1/sqrt(D)
  for (int i = lane; i < 16 * HD; i += 32) {
    const int m = i >> 6, d = i & 63;
    float2 q = qkv2[base + (size_t)(qt * 16 + m) * QKVN + h * HD + d];
    q.x *= QK_SCALE; q.y *= QK_SCALE;
    ((float2*)qs)[i] = q;
  }
  __syncthreads();

  const v8f z = {0, 0, 0, 0, 0, 0, 0, 0};
  v8f od_r[4], od_i[4];
#pragma unroll
  for (int t = 0; t < 4; ++t) { od_r[t] = z; od_i[t] = z; }
  float mrow[8], lrow[8];
#pragma unroll
  for (int v = 0; v < 8; ++v) { mrow[v] = -INFINITY; lrow[v] = 0.f; }

  const float* qrow = qs + lo * (2 * HD) + koff;
  const float2* vbase = qkv2 + base + 2 * DIM + h * HD;

  for (int kt = 0; kt < 64; ++kt) {
    // key row (tileN side) pointers for this tile, doubled-real dim = 128
    const size_t krow = (btok + (size_t)kt * 16 + lo) * (2 * DIM) + h * (2 * HD) + koff;
    const float* k1p = k1 + krow;
    const float* k2p = k2 + krow;

    v8f sr = z, si = z;
    // S = Q * K^T over 2D = 128 reals -> 32 K-steps, 2 WMMAs each (no shuffles)
#pragma unroll 8
    for (int ks = 0; ks < 32; ++ks) {
      const int k0 = ks * 4;
      const v2f a  = *(const v2f*)(qrow + k0);
      const v2f b1 = *(const v2f*)(k1p + k0);
      const v2f b2 = *(const v2f*)(k2p + k0);
      sr = wmma4(a, b1, sr);
      si = wmma4(a, b2, si);
    }

    // magnitude + online softmax (row = v + 8*hi, striped over 16-lane halves)
    float magv[8], pv[8];
#pragma unroll
    for (int v = 0; v < 8; ++v) magv[v] = sqrtf(sr[v] * sr[v] + si[v] * si[v]);
#pragma unroll
    for (int v = 0; v < 8; ++v) {
      float rm = magv[v];
      rm = fmaxf(rm, __shfl_xor(rm, 1));
      rm = fmaxf(rm, __shfl_xor(rm, 2));
      rm = fmaxf(rm, __shfl_xor(rm, 4));
      rm = fmaxf(rm, __shfl_xor(rm, 8));
      const float mnew = fmaxf(mrow[v], rm);
      const float sc = __expf(mrow[v] - mnew);
      const float p = __expf(magv[v] - mnew);
      float rsum = p;
      rsum += __shfl_xor(rsum, 1);
      rsum += __shfl_xor(rsum, 2);
      rsum += __shfl_xor(rsum, 4);
      rsum += __shfl_xor(rsum, 8);
      lrow[v] = lrow[v] * sc + rsum;
      mrow[v] = mnew;
      pv[v] = p;
#pragma unroll
      for (int t = 0; t < 4; ++t) { od_r[t][v] *= sc; od_i[t][v] *= sc; }
    }

    // route P from C-layout to A-layout through LDS
    __syncthreads();
#pragma unroll
    for (int v = 0; v < 8; ++v) ps[(v + 8 * hi) * 16 + lo] = pv[v];
    __syncthreads();

    // O += P(16x16, real) @ V(16x64, complex): 4 K-steps x 4 d-tiles x 2 WMMAs
#pragma unroll
    for (int ks2 = 0; ks2 < 4; ++ks2) {
      const int kk = ks2 * 4 + koff;
      const v2f pa = *(const v2f*)(ps + lo * 16 + kk);
      const int key0 = kt * 16 + kk;
#pragma unroll
      for (int t = 0; t < 4; ++t) {
        const int d = t * 16 + lo;
        const float2 v0 = vbase[(size_t)key0 * QKVN + d];
        const float2 v1 = vbase[(size_t)(key0 + 1) * QKVN + d];
        const v2f vr = {v0.x, v1.x}, vi = {v0.y, v1.y};
        od_r[t] = wmma4(pa, vr, od_r[t]);
        od_i[t] = wmma4(pa, vi, od_i[t]);
      }
    }
  }

  // normalize and write ctx[token, h*64 + d]
  float2* ctx2 = (float2*)ctx;
#pragma unroll
  for (int v = 0; v < 8; ++v) {
    const float inv = 1.0f / lrow[v];
    const int m = qt * 16 + v + 8 * hi;
    const size_t o = (btok + m) * DIM + h * HD;
#pragma unroll
    for (int t = 0; t < 4; ++t) {
      float2 y;
      y.x = od_r[t][v] * inv;
      y.y = od_i[t][v] * inv;
      ctx2[o + t * 16 + lo] = y;
    }
  }
}

// ---------------------------------------------------------------------------
// Host-side orchestration (graph-capture safe: kernels only, all on stream)
// ---------------------------------------------------------------------------
extern "C" void kernel_launch(void* const* d_in, const int* in_sizes, int n_in,
                              void* d_out, int out_size, void* d_ws, size_t ws_size,
                              hipStream_t stream) {
  const float* x      = (const float*)d_in[0];
  const float* n1_w   = (const float*)d_in[1];
  const float* n1_b   = (const float*)d_in[2];
  const float* qkv_w  = (const float*)d_in[3];
  const float* proj_w = (const float*)d_in[4];
  const float* proj_b = (const float*)d_in[5];
  const float* n2_w   = (const float*)d_in[6];
  const float* n2_b   = (const float*)d_in[7];
  const float* fc1_w  = (const float*)d_in[8];
  const float* fc1_b  = (const float*)d_in[9];
  const float* fc2_w  = (const float*)d_in[10];
  const float* fc2_b  = (const float*)d_in[11];
  float* out = (float*)d_out;

  float* ws = (float*)d_ws;
  float* lnb  = ws;                                   // [4096,768]cx   (ln out)
  float* big  = lnb + (size_t)NTOK * DIM * 2;         // qkv [4096,2304]cx -> hidden [4096,3072]cx
  float* ctxb = big + (size_t)NTOK * HID * 2;         // ctx [4096,768]cx
  float* kb1  = ctxb + (size_t)NTOK * DIM * 2;        // K1 [4096,768]cx
  float* kb2  = kb1 + (size_t)NTOK * DIM * 2;         // K2 [4096,768]cx
  float* wsp  = kb2 + (size_t)NTOK * DIM * 2;         // transformed weights

  const int n_qkv  = QKVN * DIM;   // complex elems
  const int n_proj = DIM * DIM;
  const int n_fc1  = HID * DIM;
  const int n_fc2  = DIM * HID;

  float* qkv_w1  = wsp;                          float* qkv_w2  = qkv_w1  + (size_t)n_qkv * 2;
  float* proj_w1 = qkv_w2  + (size_t)n_qkv * 2;  float* proj_w2 = proj_w1 + (size_t)n_proj * 2;
  float* fc1_w1  = proj_w2 + (size_t)n_proj * 2; float* fc1_w2  = fc1_w1  + (size_t)n_fc1 * 2;
  float* fc2_w1  = fc1_w2  + (size_t)n_fc1 * 2;  float* fc2_w2  = fc2_w1  + (size_t)n_fc2 * 2;

  // 0) weight transforms (W1 = (wr,-wi), W2 = (wi,wr))
  make_w12<<<dim3((n_qkv  + 255) / 256), dim3(256), 0, stream>>>(qkv_w,  qkv_w1,  qkv_w2,  n_qkv);
  make_w12<<<dim3((n_proj + 255) / 256), dim3(256), 0, stream>>>(proj_w, proj_w1, proj_w2, n_proj);
  make_w12<<<dim3((n_fc1  + 255) / 256), dim3(256), 0, stream>>>(fc1_w,  fc1_w1,  fc1_w2,  n_fc1);
  make_w12<<<dim3((n_fc2  + 255) / 256), dim3(256), 0, stream>>>(fc2_w,  fc2_w1,  fc2_w2,  n_fc2);

  // 1) ln1(x) -> lnb
  ln_cplx<<<dim3(NTOK), dim3(32), 0, stream>>>(x, n1_w, n1_b, lnb);
  // 2) qkv = lnb @ qkv_w^T
  gemm_cplx2<false, false, false><<<dim3(QKVN / 64, NTOK / 32), dim3(256), 0, stream>>>(
      lnb, qkv_w1, qkv_w2, nullptr, nullptr, big, NTOK, QKVN, 2 * DIM);
  // 2b) K section transform for doubled-real scores
  make_k12<<<dim3((NTOK * DIM) / 256), dim3(256), 0, stream>>>(big, kb1, kb2);
  // 3) attention -> ctxb
  attn_flash<<<dim3(4 * NHEAD, SEQ / 16), dim3(32), 0, stream>>>(big, kb1, kb2, ctxb);
  // 4) x1 = x + ctx @ proj_w^T + proj_b -> d_out
  gemm_cplx2<true, false, true><<<dim3(DIM / 64, NTOK / 32), dim3(256), 0, stream>>>(
      ctxb, proj_w1, proj_w2, proj_b, x, out, NTOK, DIM, 2 * DIM);
  // 5) ln2(x1) -> lnb
  ln_cplx<<<dim3(NTOK), dim3(32), 0, stream>>>(out, n2_w, n2_b, lnb);
  // 6) h = gelu(lnb @ fc1_w^T + fc1_b) -> big
  gemm_cplx2<true, true, false><<<dim3(HID / 64, NTOK / 32), dim3(256), 0, stream>>>(
      lnb, fc1_w1, fc1_w2, fc1_b, nullptr, big, NTOK, HID, 2 * DIM);
  // 7) out = x1 + h @ fc2_w^T + fc2_b (residual read+write d_out, same lane)
  gemm_cplx2<true, false, true><<<dim3(DIM / 64, NTOK / 32), dim3(256), 0, stream>>>(
      big, fc2_w1, fc2_w2, fc2_b, out, out, NTOK, DIM, 2 * HID);
}